// GATv3Module_23596550324285
// MI455X (gfx1250) — compile-verified
//
#include <hip/hip_runtime.h>
#include <math.h>

// ---------------- problem constants (match reference) ----------------
#define N_NODES 50000
#define E_EDGES 1600000
#define IN_DIM  128
#define H_HEADS 4
#define C_CH    16
#define OUT_DIM 64          // H*C
#define NEG_SLOPE 0.2f

typedef float v2f __attribute__((ext_vector_type(2)));
typedef float v8f __attribute__((ext_vector_type(8)));

// =====================================================================
// Kernel 1: xp = x @ W^T  via V_WMMA_F32_16X16X4_F32
//   x : [N, 128] row-major,  W : [64, 128] row-major, xp : [N, 64]
//   block = 128 threads = 4 waves; block handles 16 rows; wave w -> cols 16w..16w+15
//   A 16x4 f32 layout: lanes 0-15 hold M=lane, K=k..k+1; lanes 16-31 hold M=lane-16, K=k+2..k+3
//   B 4x16 layout mirrors A with N instead of M; B[k][n] = W[n][k]
//   C/D 16x16: VGPR r, lanes 0-15 -> (M=r, N=lane); lanes 16-31 -> (M=r+8, N=lane-16)
// =====================================================================
__global__ __launch_bounds__(128) void gat_gemm_xp(const float* __restrict__ x,
                                                   const float* __restrict__ W,
                                                   float* __restrict__ xp) {
    const int row0 = blockIdx.x * 16;
    const int wave = threadIdx.x >> 5;   // 0..3  (head / col tile)
    const int lane = threadIdx.x & 31;
    const int col0 = wave * 16;
    const int l15  = lane & 15;
    const int hi   = lane >> 4;          // 0 or 1 -> K subpair

    const float* aptr = x + (size_t)(row0 + l15) * IN_DIM + hi * 2;
    const float* bptr = W + (size_t)(col0 + l15) * IN_DIM + hi * 2;

    v8f acc = {};
#pragma unroll
    for (int k = 0; k < IN_DIM; k += 4) {
        v2f a = *(const v2f*)(aptr + k);
        v2f b = *(const v2f*)(bptr + k);
        acc = __builtin_amdgcn_wmma_f32_16x16x4_f32(
            /*neg_a=*/false, a, /*neg_b=*/false, b,
            /*c_mod=*/(short)0, acc, /*reuse_a=*/false, /*reuse_b=*/false);
    }

    float* o = xp + (size_t)(row0 + hi * 8) * OUT_DIM + col0 + l15;
#pragma unroll
    for (int r = 0; r < 8; ++r) {
        o[(size_t)r * OUT_DIM] = acc[r];
    }
}

// =====================================================================
// Kernel 2: per-(node,head) attention logits + init m/s/out
// =====================================================================
__global__ void gat_node_prep(const float* __restrict__ xp,
                              const float* __restrict__ att_src,
                              const float* __restrict__ att_dst,
                              const float* __restrict__ bias,
                              float* __restrict__ a_src,
                              float* __restrict__ a_dst,
                              float* __restrict__ m,
                              float* __restrict__ s,
                              float* __restrict__ out) {
    int t = blockIdx.x * blockDim.x + threadIdx.x;   // n*H + h
    if (t >= N_NODES * H_HEADS) return;
    int n = t >> 2;
    int h = t & 3;
    const float* xr = xp + (size_t)n * OUT_DIM + h * C_CH;
    const float* as = att_src + h * C_CH;
    const float* ad = att_dst + h * C_CH;
    float ssum = 0.f, dsum = 0.f;
#pragma unroll
    for (int c = 0; c < C_CH; ++c) {
        float v = xr[c];
        ssum += v * as[c];
        dsum += v * ad[c];
    }
    a_src[t] = ssum;
    a_dst[t] = dsum;
    m[t] = -INFINITY;
    s[t] = 0.f;
    float* orow = out + (size_t)n * OUT_DIM + h * C_CH;
    const float* brow = bias + h * C_CH;
#pragma unroll
    for (int c = 0; c < C_CH; ++c) orow[c] = brow[c];
}

// sign-aware float atomic max (works for negatives; no float-overload dependence)
__device__ __forceinline__ void atomic_max_f32(float* addr, float val) {
    int iv = __float_as_int(val);
    if (iv >= 0)
        atomicMax((int*)addr, iv);
    else
        atomicMin((unsigned int*)addr, (unsigned int)iv);
}

// =====================================================================
// Kernel 3: per-(edge,head) score e = leaky_relu(a_src[src]+a_dst[dst]); segment max
// =====================================================================
__global__ void gat_edge_score(const long long* __restrict__ ei,   // [2, E] int64
                               const float* __restrict__ a_src,
                               const float* __restrict__ a_dst,
                               float* __restrict__ e_ws,
                               float* __restrict__ m) {
    int t = blockIdx.x * blockDim.x + threadIdx.x;   // e*H + h
    if (t >= E_EDGES * H_HEADS) return;
    int e = t >> 2;
    int h = t & 3;
    int src = (int)ei[e];
    int dst = (int)ei[E_EDGES + e];
    float v = a_src[src * H_HEADS + h] + a_dst[dst * H_HEADS + h];
    v = (v > 0.f) ? v : v * NEG_SLOPE;
    e_ws[t] = v;
    atomic_max_f32(&m[dst * H_HEADS + h], v);
}

// =====================================================================
// Kernel 4: ex = exp(e - m[dst]); segment sum into s
// =====================================================================
__global__ void gat_edge_exp(const long long* __restrict__ ei,
                             float* __restrict__ e_ws,             // in: e, out: ex
                             const float* __restrict__ m,
                             float* __restrict__ s) {
    int t = blockIdx.x * blockDim.x + threadIdx.x;
    if (t >= E_EDGES * H_HEADS) return;
    int e = t >> 2;
    int h = t & 3;
    int dst = (int)ei[E_EDGES + e];
    float mv = m[dst * H_HEADS + h];
    if (mv == -INFINITY) mv = 0.f;                   // reference: empty-segment guard
    float ex = expf(e_ws[t] - mv);
    e_ws[t] = ex;
    atomicAdd(&s[dst * H_HEADS + h], ex);
}

// =====================================================================
// Kernel 5: out[dst, j] += (ex/(s[dst]+eps)) * xp[src, j]   (one thread per edge-element)
// =====================================================================
__global__ void gat_edge_scatter(const long long* __restrict__ ei,
                                 const float* __restrict__ e_ws,   // ex
                                 const float* __restrict__ s,
                                 const float* __restrict__ xp,
                                 float* __restrict__ out) {
    int t = blockIdx.x * blockDim.x + threadIdx.x;   // e*64 + j  (102.4M < 2^31)
    if (t >= E_EDGES * OUT_DIM) return;
    int e = t >> 6;
    int j = t & 63;
    int h = j >> 4;
    int src = (int)ei[e];
    int dst = (int)ei[E_EDGES + e];
    float alpha = e_ws[e * H_HEADS + h] / (s[dst * H_HEADS + h] + 1e-16f);
    atomicAdd(&out[(size_t)dst * OUT_DIM + j], alpha * xp[(size_t)src * OUT_DIM + j]);
}

// =====================================================================
extern "C" void kernel_launch(void* const* d_in, const int* in_sizes, int n_in,
                              void* d_out, int out_size, void* d_ws, size_t ws_size,
                              hipStream_t stream) {
    const float*     x       = (const float*)d_in[0];
    const long long* ei      = (const long long*)d_in[1];   // int64 edge_index [2,E]
    const float*     W       = (const float*)d_in[2];
    const float*     att_src = (const float*)d_in[3];
    const float*     att_dst = (const float*)d_in[4];
    const float*     bias    = (const float*)d_in[5];
    float*           out     = (float*)d_out;

    // workspace layout (all regions fully written before any read)
    char* ws = (char*)d_ws;
    float* xp    = (float*)ws; ws += sizeof(float) * (size_t)N_NODES * OUT_DIM;   // 12.8 MB
    float* a_src = (float*)ws; ws += sizeof(float) * (size_t)N_NODES * H_HEADS;   // 0.8 MB
    float* a_dst = (float*)ws; ws += sizeof(float) * (size_t)N_NODES * H_HEADS;   // 0.8 MB
    float* m     = (float*)ws; ws += sizeof(float) * (size_t)N_NODES * H_HEADS;   // 0.8 MB
    float* ssum  = (float*)ws; ws += sizeof(float) * (size_t)N_NODES * H_HEADS;   // 0.8 MB
    float* e_ws  = (float*)ws; ws += sizeof(float) * (size_t)E_EDGES * H_HEADS;   // 25.6 MB

    // 1) projection GEMM via WMMA (50000 rows = 3125 exact 16-row tiles)
    gat_gemm_xp<<<N_NODES / 16, 128, 0, stream>>>(x, W, xp);

    // 2) node-level logits + init
    int nh = N_NODES * H_HEADS;
    gat_node_prep<<<(nh + 255) / 256, 256, 0, stream>>>(xp, att_src, att_dst, bias,
                                                        a_src, a_dst, m, ssum, out);

    // 3) edge scores + segment max
    int eh = E_EDGES * H_HEADS;
    gat_edge_score<<<(eh + 255) / 256, 256, 0, stream>>>(ei, a_src, a_dst, e_ws, m);

    // 4) exp + segment sum
    gat_edge_exp<<<(eh + 255) / 256, 256, 0, stream>>>(ei, e_ws, m, ssum);

    // 5) weighted scatter-add of messages
    int total = E_EDGES * OUT_DIM;
    gat_edge_scatter<<<(total + 255) / 256, 256, 0, stream>>>(ei, e_ws, ssum, xp, out);
}